// MultiHeadConvAttention_14379550507781
// MI455X (gfx1250) — compile-verified
//
#include <hip/hip_runtime.h>
#include <math.h>

// CDNA5 / gfx1250: wave32, V_WMMA_F32_16X16X4_F32 (full-precision f32 matrix op)
typedef __attribute__((ext_vector_type(2))) float v2f;
typedef __attribute__((ext_vector_type(8))) float v8f;

namespace {
constexpr int kB = 2, kLQ = 24, kLK = 24, kNH = 4, kDM = 32, kDF = 32, kHW = 256;
constexpr int kNC = kNH * kDM;          // 128 merged head-channels
constexpr int kImgs = kB * kLQ;         // 48 (b,l) images
constexpr float kNegSlope = 0.1f;
}

// Y[img][o][p] = act( sum_c W[o][c] * X[img][c][p] (+ bias[o]) )
// X: [nimg][K][256] f32, W: [M][K] row-major, Y: [nimg][M][256]
// One 16(M)x16(pixels) f32 tile per wave, K stepped by 4 via WMMA f32 16x16x4.
template <int M, int K, bool LRELU, bool BIAS>
__global__ __launch_bounds__(256) void gemm_wmma_k(const float* __restrict__ X,
                                                   const float* __restrict__ W,
                                                   const float* __restrict__ bias,
                                                   float* __restrict__ Y, int nimg) {
  __shared__ float sW[M * K];
  __shared__ float sB[M];
  for (int i = threadIdx.x; i < M * K; i += 256) sW[i] = W[i];
  if constexpr (BIAS) {
    for (int i = threadIdx.x; i < M; i += 256) sB[i] = bias[i];
  }
  __syncthreads();

  constexpr int MT = M / 16;
  constexpr int NT = kHW / 16;
  const int wave = (blockIdx.x * 256 + (int)threadIdx.x) >> 5;
  const int tiles = nimg * MT * NT;
  if (wave >= tiles) return;  // wave-uniform; grids are exact anyway (EXEC all-1 for WMMA)

  const int img = wave / (MT * NT);
  const int rem = wave - img * (MT * NT);
  const int o0 = (rem / NT) * 16;
  const int p0 = (rem % NT) * 16;

  const int lane = threadIdx.x & 31;
  const int lm = lane & 15;  // M (A) / N (B,C,D) index inside tile
  const int hi = lane >> 4;  // lane half selects K pair (A/B) and M+8 (C/D)

  const float* __restrict__ Xi = X + (size_t)img * K * kHW + p0 + lm;
  const float* __restrict__ Wr = sW + (o0 + lm) * K;

  v8f acc = {};
#pragma unroll
  for (int kk = 0; kk < K; kk += 4) {
    // A 16x4 f32 layout: v0 = K=kk+2*hi, v1 = K=kk+2*hi+1, row M = lane&15
    v2f a, b;
    a.x = Wr[kk + 2 * hi + 0];
    a.y = Wr[kk + 2 * hi + 1];
    // B 4x16 f32 layout (mirrored): col N = lane&15, rows kk+2*hi, kk+2*hi+1
    b.x = Xi[(kk + 2 * hi + 0) * kHW];
    b.y = Xi[(kk + 2 * hi + 1) * kHW];
    // (neg_a, A, neg_b, B, c_mod, C, reuse_a, reuse_b)
    acc = __builtin_amdgcn_wmma_f32_16x16x4_f32(false, a, false, b, (short)0, acc,
                                                false, false);
  }

  // C/D layout: VGPR r -> M = o0 + r + 8*hi, N = p0 + (lane&15)
  float* __restrict__ Yi = Y + (size_t)img * M * kHW + p0 + lm;
#pragma unroll
  for (int r = 0; r < 8; ++r) {
    const int o = o0 + r + hi * 8;
    float val = acc[r];
    if constexpr (BIAS) val += sB[o];
    if constexpr (LRELU) val = (val >= 0.f) ? val : kNegSlope * val;
    Yi[o * kHW] = val;
  }
}

// 5x5 conv, 32 in-ch -> 1 out-ch, pad 2. One block per (b,n,l) image of q or k.
// Linearity trick: conv(q_i + k_j) + b = conv(q_i) + (conv(k_j) + b).
__global__ __launch_bounds__(256) void conv5x5_k(const float* __restrict__ P,
                                                 const float* __restrict__ wat,
                                                 const float* __restrict__ battn,
                                                 float* __restrict__ Lout) {
  __shared__ float sx[kDM * kHW];  // 32 KB image tile (32 ch x 16x16)
  __shared__ float sw[kDM * 25];
  const int blk = blockIdx.x;  // (b*NH + n)*LQ + l
  const int l = blk % kLQ;
  const int n = (blk / kLQ) % kNH;
  const int b = blk / (kLQ * kNH);
  const float* __restrict__ src = P + (((size_t)(b * kLQ + l)) * kNC + n * kDM) * kHW;
  for (int i = threadIdx.x; i < kDM * kHW; i += 256) sx[i] = src[i];
  for (int i = threadIdx.x; i < kDM * 25; i += 256) sw[i] = wat[i];
  __syncthreads();

  const int p = threadIdx.x;
  const int h = p >> 4, w = p & 15;
  float acc = battn ? battn[0] : 0.f;
  for (int c = 0; c < kDM; ++c) {
    const float* __restrict__ xc = sx + c * kHW;
    const float* __restrict__ wc = sw + c * 25;
#pragma unroll
    for (int dy = -2; dy <= 2; ++dy) {
      const int hh = h + dy;
      if ((unsigned)hh >= 16u) continue;
#pragma unroll
      for (int dx = -2; dx <= 2; ++dx) {
        const int ww = w + dx;
        if ((unsigned)ww >= 16u) continue;
        acc = fmaf(xc[hh * 16 + ww], wc[(dy + 2) * 5 + (dx + 2)], acc);
      }
    }
  }
  Lout[(size_t)blk * kHW + p] = acc;
}

// attn[b,n,i,j,p] = softmax_j( Lq[b,n,i,p] + Lk[b,n,j,p] ); one lane per (b,n,i,p)
__global__ __launch_bounds__(256) void softmax_k(const float* __restrict__ Lq,
                                                 const float* __restrict__ Lk,
                                                 float* __restrict__ attn) {
  const int t = blockIdx.x * 256 + threadIdx.x;  // over B*NH*LQ*HW
  const int p = t & (kHW - 1);
  const int i = (t >> 8) % kLQ;
  const int bn = (t >> 8) / kLQ;

  const float lq = Lq[((size_t)bn * kLQ + i) * kHW + p];
  const float* __restrict__ lkp = Lk + (size_t)bn * kLK * kHW + p;

  float tv[kLK];
  float m = -3.402823466e38f;
#pragma unroll
  for (int j = 0; j < kLK; ++j) {
    const float x = lq + lkp[j * kHW];
    tv[j] = x;
    m = fmaxf(m, x);
  }
  float s = 0.f;
#pragma unroll
  for (int j = 0; j < kLK; ++j) {
    const float e = __expf(tv[j] - m);
    tv[j] = e;
    s += e;
  }
  const float inv = 1.f / s;
  float* __restrict__ ap = attn + ((size_t)(bn * kLQ + i)) * kLK * kHW + p;
#pragma unroll
  for (int j = 0; j < kLK; ++j) ap[j * kHW] = tv[j] * inv;
}

// res[b][q][n*32+c][p] = sum_k attn[b,n,q,k,p] * vh[b][k][n*32+c][p]
// Batched-per-pixel tiny matmul: gather pattern unsuitable for WMMA lane layout,
// so keep it on VALU FMAs. 2 queries per block to bound register pressure.
__global__ __launch_bounds__(256) void attnv_k(const float* __restrict__ attn,
                                               const float* __restrict__ vh,
                                               float* __restrict__ res) {
  const int blk = blockIdx.x;  // B*NH*(LQ/2)
  const int qt = blk % (kLQ / 2);
  const int n = (blk / (kLQ / 2)) % kNH;
  const int b = blk / ((kLQ / 2) * kNH);
  const int p = threadIdx.x;
  const int q0 = qt * 2;

  float acc0[kDM], acc1[kDM];
#pragma unroll
  for (int c = 0; c < kDM; ++c) { acc0[c] = 0.f; acc1[c] = 0.f; }

  const float* __restrict__ ap =
      attn + ((size_t)((b * kNH + n) * kLQ + q0)) * kLK * kHW + p;
  for (int k = 0; k < kLK; ++k) {
    const float a0 = ap[k * kHW];
    const float a1 = ap[(kLK + k) * kHW];
    const float* __restrict__ vk =
        vh + (((size_t)(b * kLK + k)) * kNC + n * kDM) * kHW + p;
#pragma unroll
    for (int c = 0; c < kDM; ++c) {
      const float vv = vk[c * kHW];
      acc0[c] = fmaf(a0, vv, acc0[c]);
      acc1[c] = fmaf(a1, vv, acc1[c]);
    }
  }
  float* __restrict__ r0 = res + (((size_t)(b * kLQ + q0)) * kNC + n * kDM) * kHW + p;
  float* __restrict__ r1 = r0 + (size_t)kNC * kHW;
#pragma unroll
  for (int c = 0; c < kDM; ++c) {
    r0[c * kHW] = acc0[c];
    r1[c * kHW] = acc1[c];
  }
}

extern "C" void kernel_launch(void* const* d_in, const int* in_sizes, int n_in,
                              void* d_out, int out_size, void* d_ws, size_t ws_size,
                              hipStream_t stream) {
  const float* q = (const float*)d_in[0];
  const float* k = (const float*)d_in[1];
  const float* v = (const float*)d_in[2];
  const float* wq = (const float*)d_in[3];
  const float* bq = (const float*)d_in[4];
  const float* wk = (const float*)d_in[5];
  const float* bk = (const float*)d_in[6];
  const float* wv = (const float*)d_in[7];
  const float* bv = (const float*)d_in[8];
  const float* watt = (const float*)d_in[9];   // (1,32,5,5)
  const float* batt = (const float*)d_in[10];  // (1,)
  const float* wfc = (const float*)d_in[11];   // (32,128)

  // workspace layout (floats): qh | kh | vh | Lq | Lk ; res reuses qh (~19.3 MB)
  float* ws = (float*)d_ws;
  const size_t projN = (size_t)kImgs * kNC * kHW;  // 1,572,864
  float* qh = ws;
  float* kh = qh + projN;
  float* vh = kh + projN;
  float* Lq = vh + projN;
  float* Lk = Lq + (size_t)kB * kNH * kLQ * kHW;
  float* res = qh;  // qh dead after conv5x5_k(qh)

  float* out = (float*)d_out;                        // (B,LQ,32,16,16)
  float* attn = out + (size_t)kB * kLQ * kDM * kHW;  // (B,NH,LQ,LK,1,16,16)

  // 1) projections: WMMA f32 GEMM, 48 imgs * 8 o-tiles * 16 p-tiles = 6144 waves
  gemm_wmma_k<kNC, kDF, true, true><<<768, 256, 0, stream>>>(q, wq, bq, qh, kImgs);
  gemm_wmma_k<kNC, kDF, true, true><<<768, 256, 0, stream>>>(k, wk, bk, kh, kImgs);
  gemm_wmma_k<kNC, kDF, true, true><<<768, 256, 0, stream>>>(v, wv, bv, vh, kImgs);

  // 2) linearized attention conv: Lq = conv(qh), Lk = conv(kh) + b_attn
  conv5x5_k<<<kB * kNH * kLQ, 256, 0, stream>>>(qh, watt, nullptr, Lq);
  conv5x5_k<<<kB * kNH * kLK, 256, 0, stream>>>(kh, watt, batt, Lk);

  // 3) softmax over keys -> attn (directly into d_out slice)
  softmax_k<<<(kB * kNH * kLQ * kHW) / 256, 256, 0, stream>>>(Lq, Lk, attn);

  // 4) weighted sum over keys, merged-head layout
  attnv_k<<<kB * kNH * (kLQ / 2), 256, 0, stream>>>(attn, vh, res);

  // 5) final fc: WMMA f32 GEMM, 48 * 2 * 16 = 1536 waves
  gemm_wmma_k<kDM, kNC, false, false><<<192, 256, 0, stream>>>(res, wfc, nullptr, out,
                                                               kImgs);
}